// SimulateQuantizedDense_57286273794900
// MI455X (gfx1250) — compile-verified
//
#include <hip/hip_runtime.h>
#include <hip/hip_fp16.h>
#include <stdint.h>

// ---------------------------------------------------------------------------
// Fused Q4_0 fake-quant dequant + (32x8192)x(8192x8192) GEMM + bias, MI455X.
//
// Memory-bound (W = 256MB f32 @ 23.3 TB/s => ~11us floor):
//   * Weights stream HBM -> LDS via the Tensor Data Mover (tensor_load_to_lds,
//     one 4KB 16x64 descriptor per super-tile, double buffered, TENSORcnt).
//   * LDS tile is read twice: quarter-block-per-lane layout for Q4_0 stats
//     (in-lane max/min tree + 2-step xor butterfly), and B-matrix layout for
//     the V_WMMA_F32_16X16X4_F32 accumulation (full f32 = reference math).
//   * Scale math (fp16 round-trip, rcp+Newton) is lane-parallel: once per
//     K-step for all 8 blocks.
// Partition: 128 N-strips x 16 K-splits, one wave32 per workgroup; partials
// to d_ws, deterministic fixed-order reduction + bias kernel.
// ---------------------------------------------------------------------------

typedef __attribute__((ext_vector_type(2))) float v2f;
typedef __attribute__((ext_vector_type(4))) float v4f;
typedef __attribute__((ext_vector_type(8))) float v8f;
typedef __attribute__((ext_vector_type(4))) unsigned int v4u;
typedef __attribute__((ext_vector_type(4))) int v4i;
typedef __attribute__((ext_vector_type(8))) int v8i;
typedef unsigned int u32;
typedef unsigned long long u64;

#define D_IN   8192
#define FEATS  8192
#define M_ROWS 32
#define KSPLIT 16
#define KSEG   (D_IN / KSPLIT)    // 512
#define NSTRIP 64                 // N columns per wave
#define SUPER  16                 // K rows per TDM tile (4 WMMA K-steps)
#define NSUP   (KSEG / SUPER)     // 32
#define BUF_ELTS (SUPER * NSTRIP) // 1024 floats = 4KB per buffer

// Tensor DMA descriptor (D#) for a 2-D 64x16 f32 tile of W (row stride 8192).
__device__ __forceinline__ void tdm_load_tile(const float* gsrc, u32 lds_off) {
    u64 ga = (u64)(uintptr_t)gsrc;
    v4u g0;
    g0.x = 1u;                                            // count=1 (valid)
    g0.y = lds_off;                                       // lds_addr (bytes)
    g0.z = (u32)ga;                                       // global_addr[31:0]
    g0.w = (u32)((ga >> 32) & 0x01FFFFFFu) | (2u << 30);  // addr[56:32] | type=2
    v8i g1;
    g1[0] = 0x00020000;                        // workgroup_mask=0, data_size=4B
    g1[1] = (int)(8192u << 16);                // tensor_dim0[15:0]<<16
    g1[2] = (int)((8192u >> 16) | (8192u << 16)); // dim0 hi | tensor_dim1 lo
    g1[3] = (int)(64u << 16);                  // dim1 hi | tile_dim0=64
    g1[4] = SUPER;                             // tile_dim1=16, tile_dim2=0
    g1[5] = 8192;                              // tensor_dim0_stride lo32
    g1[6] = 0;                                 // stride hi | dim1_stride lo
    g1[7] = 0;
    v4i z4 = {0, 0, 0, 0};                     // groups 2/3 unused (2-D tile)
    v8i z8 = {0, 0, 0, 0, 0, 0, 0, 0};
    __builtin_amdgcn_tensor_load_to_lds(g0, g1, z4, z4, z8, 0);
}

__device__ __forceinline__ float dq(float x, float inv, float scale, float s8n) {
    float q = truncf(fmaf(x, inv, 8.5f));      // trunc(x/d + 8.5)
    q = fminf(fmaxf(q, 0.0f), 15.0f);          // -> v_med3
    return fmaf(q, scale, s8n);                // scale*(q-8)
}

__global__ __launch_bounds__(32)
void q4gemm_kernel(const float* __restrict__ x,   // [32][8192]
                   const float* __restrict__ w,   // [8192][8192]
                   float* __restrict__ part)      // [KSPLIT][32][8192]
{
    __shared__ __align__(16) float lbuf[2 * BUF_ELTS];   // 8KB double buffer

    const int lane  = threadIdx.x;         // 0..31
    const int n0    = blockIdx.x * NSTRIP;
    const int kseg  = blockIdx.y;
    const int kbase = kseg * KSEG;
    const int lhalf = lane >> 4;
    const int l15   = lane & 15;

    // Never-taken store: marks LDS as written so TDM-filled reads can't fold.
    if ((int)blockIdx.x < 0) lbuf[lane] = 0.0f;

    // Stats layout: lane owns a quarter-block (8 floats); block = (row<<1)|nhalf
    const int blk      = lane >> 2;                       // 0..7
    const int stat_off = (blk >> 1) * NSTRIP + (blk & 1) * 32 + (lane & 3) * 8;

    // Broadcast source lanes: block b' stats live in lanes 4b'..4b'+3
    const int sl00 = lhalf * 16;        // (v=0, nhalf=0)
    const int sl01 = sl00 + 4;          // (v=0, nhalf=1)
    const int sl10 = sl00 + 8;          // (v=1, nhalf=0)
    const int sl11 = sl00 + 12;         // (v=1, nhalf=1)

    // A layout (16x4 f32): VGPR0 = K{0|2}, VGPR1 = K{1|3}; M = lane&15
    const float* xa = x + (size_t)l15 * D_IN + kbase + lhalf * 2;
    const float* wtile = w + (size_t)kbase * FEATS + n0;
    const u32 lds_base = (u32)(uintptr_t)&lbuf[0];        // low 32 bits = LDS offset

    v8f acc[2][4];
#pragma unroll
    for (int mi = 0; mi < 2; ++mi)
#pragma unroll
        for (int j = 0; j < 4; ++j) acc[mi][j] = {};

    tdm_load_tile(wtile, lds_base);                       // super 0 -> buf 0

#pragma unroll 1
    for (int s = 0; s < NSUP; ++s) {
        if (s + 1 < NSUP) {
            // Next buffer was read by super s-1: make sure those ds reads are done
            asm volatile("s_wait_dscnt 0x0" ::: "memory");
            tdm_load_tile(wtile + (size_t)(s + 1) * SUPER * FEATS,
                          lds_base + (u32)(((s + 1) & 1) * BUF_ELTS * 4));
            __builtin_amdgcn_s_wait_tensorcnt(1);         // current buffer ready
        } else {
            __builtin_amdgcn_s_wait_tensorcnt(0);
        }
        asm volatile("" ::: "memory");

        const float* B = lbuf + (s & 1) * BUF_ELTS;

#pragma unroll
        for (int t = 0; t < 4; ++t) {
            const float* Bt = B + t * (4 * NSTRIP);

            // ---- Q4_0 stats: in-lane tree over 8 floats, then 4-lane butterfly
            v4f s0 = *(const v4f*)(Bt + stat_off);
            v4f s1 = *(const v4f*)(Bt + stat_off + 4);
            float mx = fmaxf(fmaxf(fmaxf(s0.x, s0.y), fmaxf(s0.z, s0.w)),
                             fmaxf(fmaxf(s1.x, s1.y), fmaxf(s1.z, s1.w)));
            float mn = fminf(fminf(fminf(s0.x, s0.y), fminf(s0.z, s0.w)),
                             fminf(fminf(s1.x, s1.y), fminf(s1.z, s1.w)));
            mx = fmaxf(mx, __shfl_xor(mx, 1, 32));
            mn = fminf(mn, __shfl_xor(mn, 1, 32));
            mx = fmaxf(mx, __shfl_xor(mx, 2, 32));
            mn = fminf(mn, __shfl_xor(mn, 2, 32));

            float val   = (-mn > mx) ? mn : mx;           // signed abs-argmax value
            float d     = val * -0.125f;                  // -max/8
            float scale = __half2float(__float2half(d));  // fp16 round-trip
            float inv   = __builtin_amdgcn_rcpf(d);       // 1/d (+Newton)
            inv = fmaf(inv, fmaf(-d, inv, 1.0f), inv);
            inv = (d == 0.0f) ? 0.0f : inv;               // safe_inv_d

            // ---- broadcast per-block (inv, scale) to B-layout lanes
            float i00 = __shfl(inv, sl00, 32), c00 = __shfl(scale, sl00, 32);
            float i01 = __shfl(inv, sl01, 32), c01 = __shfl(scale, sl01, 32);
            float i10 = __shfl(inv, sl10, 32), c10 = __shfl(scale, sl10, 32);
            float i11 = __shfl(inv, sl11, 32), c11 = __shfl(scale, sl11, 32);
            float n00 = c00 * -8.0f, n01 = c01 * -8.0f;
            float n10 = c10 * -8.0f, n11 = c11 * -8.0f;

            // ---- B-matrix layout reads (merge to ds_load_2addr pairs)
            const float* Bv0 = Bt + 2 * lhalf * NSTRIP + l15;
            const float* Bv1 = Bv0 + NSTRIP;
            float b00 = Bv0[0], b10 = Bv0[16], b20 = Bv0[32], b30 = Bv0[48];
            float b01 = Bv1[0], b11 = Bv1[16], b21 = Bv1[32], b31 = Bv1[48];

            b00 = dq(b00, i00, c00, n00);  b10 = dq(b10, i00, c00, n00);
            b20 = dq(b20, i01, c01, n01);  b30 = dq(b30, i01, c01, n01);
            b01 = dq(b01, i10, c10, n10);  b11 = dq(b11, i10, c10, n10);
            b21 = dq(b21, i11, c11, n11);  b31 = dq(b31, i11, c11, n11);

            // ---- A tiles + 8 f32 WMMAs
            v2f a0 = *(const v2f*)(xa + s * SUPER + 4 * t);              // M 0..15
            v2f a1 = *(const v2f*)(xa + 16 * D_IN + s * SUPER + 4 * t);  // M 16..31
            v2f B0 = {b00, b01};
            v2f B1 = {b10, b11};
            v2f B2 = {b20, b21};
            v2f B3 = {b30, b31};

            acc[0][0] = __builtin_amdgcn_wmma_f32_16x16x4_f32(false, a0, false, B0, (short)0, acc[0][0], false, false);
            acc[0][1] = __builtin_amdgcn_wmma_f32_16x16x4_f32(false, a0, false, B1, (short)0, acc[0][1], false, false);
            acc[0][2] = __builtin_amdgcn_wmma_f32_16x16x4_f32(false, a0, false, B2, (short)0, acc[0][2], false, false);
            acc[0][3] = __builtin_amdgcn_wmma_f32_16x16x4_f32(false, a0, false, B3, (short)0, acc[0][3], false, false);
            acc[1][0] = __builtin_amdgcn_wmma_f32_16x16x4_f32(false, a1, false, B0, (short)0, acc[1][0], false, false);
            acc[1][1] = __builtin_amdgcn_wmma_f32_16x16x4_f32(false, a1, false, B1, (short)0, acc[1][1], false, false);
            acc[1][2] = __builtin_amdgcn_wmma_f32_16x16x4_f32(false, a1, false, B2, (short)0, acc[1][2], false, false);
            acc[1][3] = __builtin_amdgcn_wmma_f32_16x16x4_f32(false, a1, false, B3, (short)0, acc[1][3], false, false);
        }
    }

    // C/D layout: VGPR p -> M = p (lanes 0-15) / p+8 (lanes 16-31), N = lane&15
    float* pout = part + (size_t)kseg * (M_ROWS * (size_t)FEATS);
#pragma unroll
    for (int mi = 0; mi < 2; ++mi) {
#pragma unroll
        for (int j = 0; j < 4; ++j) {
#pragma unroll
            for (int p = 0; p < 8; ++p) {
                int m = mi * 16 + p + (lhalf ? 8 : 0);
                pout[(size_t)m * FEATS + n0 + 16 * j + l15] = acc[mi][j][p];
            }
        }
    }
}

// out = bias + sum over K-splits (fixed order -> deterministic)
__global__ __launch_bounds__(256)
void reduce_bias_kernel(const float* __restrict__ part,
                        const float* __restrict__ bias,
                        float* __restrict__ out)
{
    int idx = blockIdx.x * blockDim.x + threadIdx.x;
    if (idx >= M_ROWS * FEATS) return;
    float a = bias[idx & (FEATS - 1)];
#pragma unroll
    for (int s = 0; s < KSPLIT; ++s)
        a += part[(size_t)s * (M_ROWS * (size_t)FEATS) + idx];
    out[idx] = a;
}

extern "C" void kernel_launch(void* const* d_in, const int* in_sizes, int n_in,
                              void* d_out, int out_size, void* d_ws, size_t ws_size,
                              hipStream_t stream) {
    const float* x    = (const float*)d_in[0];   // inputs [1,32,8192]
    const float* w    = (const float*)d_in[1];   // kernel [8192,8192]
    const float* bias = (const float*)d_in[2];   // bias   [8192]
    float* out  = (float*)d_out;                 // [1,32,8192]
    float* part = (float*)d_ws;                  // KSPLIT*32*8192*4 = 16 MB

    dim3 grid(FEATS / NSTRIP, KSPLIT);           // 128 x 16 workgroups
    q4gemm_kernel<<<grid, 32, 0, stream>>>(x, w, part);

    int total = M_ROWS * FEATS;
    reduce_bias_kernel<<<(total + 255) / 256, 256, 0, stream>>>(part, bias, out);
}